// GAT_84507776516708
// MI455X (gfx1250) — compile-verified
//
#include <hip/hip_runtime.h>

typedef __attribute__((ext_vector_type(2))) float v2f;
typedef __attribute__((ext_vector_type(8))) float v8f;

#define NEG_SLOPE 0.2f

// ---------- helpers ----------

// order-preserving float<->uint mapping so we can use global_atomic_max_u32
__device__ __forceinline__ unsigned f2ord(float f) {
  unsigned u = __float_as_uint(f);
  return (u & 0x80000000u) ? ~u : (u | 0x80000000u);
}
__device__ __forceinline__ float ord2f(unsigned o) {
  unsigned u = (o & 0x80000000u) ? (o & 0x7fffffffu) : ~o;
  return __uint_as_float(u);
}

__device__ __forceinline__ void edge_ends(const int* __restrict__ ei, int e,
                                          int Eraw, int& s, int& d) {
  if (e < Eraw) { s = ei[e]; d = ei[Eraw + e]; }
  else          { s = e - Eraw; d = s; }       // appended self-loops
}

// ---------- fill kernels (re-init accumulators every call) ----------

__global__ void fill_f32(float* __restrict__ p, float v, long long n) {
  long long i = (long long)blockIdx.x * blockDim.x + threadIdx.x;
  if (i < n) p[i] = v;
}
__global__ void fill_u32(unsigned* __restrict__ p, unsigned v, long long n) {
  long long i = (long long)blockIdx.x * blockDim.x + threadIdx.x;
  if (i < n) p[i] = v;
}

// ---------- fp32 WMMA GEMM: C[M,Nn] = A[M,K] @ B[K,Nn], all row-major ----------
// One wave computes a 64x16 C block (4 stacked 16x16 tiles) so each B fragment
// is reused by 4 V_WMMA_F32_16X16X4_F32 ops. Requires Nn % 16 == 0 and
// M % 16 == 0; M-tile count need not be a multiple of 4 (rows clamped).

__global__ void gemm_wmma_f32(const float* __restrict__ A,
                              const float* __restrict__ B,
                              float* __restrict__ C,
                              int M, int Nn, int K) {
  const int lane   = threadIdx.x & 31;
  const int wave   = blockIdx.x * (blockDim.x >> 5) + (threadIdx.x >> 5);
  const int ntiles = Nn >> 4;
  const int mtiles = M >> 4;                 // 16-row tiles
  const int mt64   = (mtiles + 3) >> 2;      // 64-row blocks
  const int tileM  = wave / ntiles;          // uniform within wave
  const int tileN  = wave - tileM * ntiles;
  if (tileM >= mt64) return;                 // wave-uniform exit (EXEC stays all-1s)

  const int lh  = lane >> 4;                 // 0 -> K=0,1 ; 1 -> K=2,3
  const int l16 = lane & 15;
  const int m0  = tileM * 4;                 // first 16-row subtile

  // clamp rows so the (single) remainder block loads in-bounds without branches
  const float* arow[4];
#pragma unroll
  for (int r = 0; r < 4; ++r) {
    int mr = m0 + r; if (mr > mtiles - 1) mr = mtiles - 1;
    arow[r] = A + (size_t)(mr * 16 + l16) * K + lh * 2;
  }
  const float* bptr = B + (size_t)(lh * 2) * Nn + tileN * 16 + l16;

  v8f acc0 = {}, acc1 = {}, acc2 = {}, acc3 = {};
  for (int k = 0; k < K; k += 4) {
    v2f b;
    b.x = bptr[(size_t)k * Nn];              // B[k+lh*2  ][n]
    b.y = bptr[(size_t)(k + 1) * Nn];        // B[k+lh*2+1][n]
    v2f a0 = *(const v2f*)(arow[0] + k);
    v2f a1 = *(const v2f*)(arow[1] + k);
    v2f a2 = *(const v2f*)(arow[2] + k);
    v2f a3 = *(const v2f*)(arow[3] + k);
    acc0 = __builtin_amdgcn_wmma_f32_16x16x4_f32(false, a0, false, b, (short)0, acc0, false, false);
    acc1 = __builtin_amdgcn_wmma_f32_16x16x4_f32(false, a1, false, b, (short)0, acc1, false, false);
    acc2 = __builtin_amdgcn_wmma_f32_16x16x4_f32(false, a2, false, b, (short)0, acc2, false, false);
    acc3 = __builtin_amdgcn_wmma_f32_16x16x4_f32(false, a3, false, b, (short)0, acc3, false, false);
  }

  // C/D layout: VGPR r -> M = r (lanes 0-15) / M = 8+r (lanes 16-31), N = lane&15
  const int col = tileN * 16 + l16;
  v8f accs[4] = {acc0, acc1, acc2, acc3};
#pragma unroll
  for (int r = 0; r < 4; ++r) {
    if (m0 + r < mtiles) {                   // wave-uniform guard
      float* crow = C + (size_t)((m0 + r) * 16 + lh * 8) * Nn + col;
#pragma unroll
      for (int q = 0; q < 8; ++q) crow[(size_t)q * Nn] = accs[r][q];
    }
  }
}

// ---------- per-node attention coefficients ----------

template <int H, int F>
__global__ void node_alpha(const float* __restrict__ h,
                           const float* __restrict__ att_s,
                           const float* __restrict__ att_d,
                           float* __restrict__ asrc, float* __restrict__ adst,
                           int n_nodes) {
  int idx = blockIdx.x * blockDim.x + threadIdx.x;   // node*H + head
  if (idx >= n_nodes * H) return;
  int n  = idx / H;
  int hd = idx - n * H;
  const float* hp = h + (size_t)n * (H * F) + hd * F;
  const float* as = att_s + hd * F;
  const float* ad = att_d + hd * F;
  float s = 0.f, d = 0.f;
#pragma unroll 8
  for (int f = 0; f < F; ++f) { float v = hp[f]; s += v * as[f]; d += v * ad[f]; }
  asrc[idx] = s;
  adst[idx] = d;
}

// ---------- edge passes ----------

template <int H>
__global__ void edge_max_k(const int* __restrict__ ei,
                           const float* __restrict__ asrc,
                           const float* __restrict__ adst,
                           unsigned* __restrict__ mord, int Eraw, int Etot) {
  int idx = blockIdx.x * blockDim.x + threadIdx.x;   // edge*H + head
  if (idx >= Etot * H) return;
  int e  = idx / H;
  int hd = idx - e * H;
  int s, d; edge_ends(ei, e, Eraw, s, d);
  float v = asrc[s * H + hd] + adst[d * H + hd];
  v = v > 0.f ? v : NEG_SLOPE * v;
  atomicMax(mord + d * H + hd, f2ord(v));
}

template <int H>
__global__ void edge_expsum_k(const int* __restrict__ ei,
                              const float* __restrict__ asrc,
                              const float* __restrict__ adst,
                              const unsigned* __restrict__ mord,
                              float* __restrict__ denom, int Eraw, int Etot) {
  int idx = blockIdx.x * blockDim.x + threadIdx.x;
  if (idx >= Etot * H) return;
  int e  = idx / H;
  int hd = idx - e * H;
  int s, d; edge_ends(ei, e, Eraw, s, d);
  float v = asrc[s * H + hd] + adst[d * H + hd];
  v = v > 0.f ? v : NEG_SLOPE * v;
  float m = ord2f(mord[d * H + hd]);
  atomicAdd(denom + d * H + hd, __expf(v - m));
}

// one thread per (edge, 4-channel group): float4 gather + 4 scatter atomics
template <int H, int F>
__global__ void edge_agg_k(const int* __restrict__ ei,
                           const float* __restrict__ asrc,
                           const float* __restrict__ adst,
                           const unsigned* __restrict__ mord,
                           const float* __restrict__ denom,
                           const float* __restrict__ feat,
                           float* __restrict__ out, int Eraw, int Etot) {
  const int C  = H * F;
  const int CV = C / 4;
  long long idx = (long long)blockIdx.x * blockDim.x + threadIdx.x; // edge*CV + c4
  if (idx >= (long long)Etot * CV) return;
  int e  = (int)(idx / CV);
  int c4 = (int)(idx - (long long)e * CV);
  int c  = c4 * 4;
  int hd = c / F;                              // F % 4 == 0, so group stays in one head
  int s, d; edge_ends(ei, e, Eraw, s, d);
  float v = asrc[s * H + hd] + adst[d * H + hd];
  v = v > 0.f ? v : NEG_SLOPE * v;
  float m     = ord2f(mord[d * H + hd]);
  float alpha = __expf(v - m) / (denom[d * H + hd] + 1e-16f);
  const float4 f4 = *(const float4*)(feat + (size_t)s * C + c);
  float* op = out + (size_t)d * C + c;
  atomicAdd(op + 0, f4.x * alpha);
  atomicAdd(op + 1, f4.y * alpha);
  atomicAdd(op + 2, f4.z * alpha);
  atomicAdd(op + 3, f4.w * alpha);
}

// ---------- epilogues ----------

__global__ void bias_elu(float* __restrict__ x, const float* __restrict__ b,
                         int C, long long n) {
  long long i = (long long)blockIdx.x * blockDim.x + threadIdx.x;
  if (i >= n) return;
  float v = x[i] + b[(int)(i % C)];
  x[i] = v > 0.f ? v : expm1f(v);
}

__global__ void bias_add(float* __restrict__ x, const float* __restrict__ b,
                         int C, long long n) {
  long long i = (long long)blockIdx.x * blockDim.x + threadIdx.x;
  if (i >= n) return;
  x[i] += b[(int)(i % C)];
}

// ---------- launcher ----------

static inline int cdiv(long long a, long long b) { return (int)((a + b - 1) / b); }

extern "C" void kernel_launch(void* const* d_in, const int* in_sizes, int n_in,
                              void* d_out, int out_size, void* d_ws, size_t ws_size,
                              hipStream_t stream) {
  const float* x      = (const float*)d_in[0];
  const int*   ei     = (const int*)  d_in[1];
  const float* W1     = (const float*)d_in[2];
  const float* att_s1 = (const float*)d_in[3];
  const float* att_d1 = (const float*)d_in[4];
  const float* b1     = (const float*)d_in[5];
  const float* W2     = (const float*)d_in[6];
  const float* att_s2 = (const float*)d_in[7];
  const float* att_d2 = (const float*)d_in[8];
  const float* b2     = (const float*)d_in[9];

  const int IN   = 128;
  const int C1   = 256;                  // 4 heads * 64
  const int F2   = 64;
  const int Nn   = in_sizes[0] / IN;     // 50000
  const int Eraw = in_sizes[1] / 2;      // 800000
  const int Etot = Eraw + Nn;            // 850000

  // workspace layout (floats)
  float* ws = (float*)d_ws;
  size_t off = 0;
  float*    h1     = ws + off; off += (size_t)Nn * C1;
  float*    out1   = ws + off; off += (size_t)Nn * C1;
  float*    asrc1  = ws + off; off += (size_t)Nn * 4;
  float*    adst1  = ws + off; off += (size_t)Nn * 4;
  unsigned* mord1  = (unsigned*)(ws + off); off += (size_t)Nn * 4;
  float*    denom1 = ws + off; off += (size_t)Nn * 4;
  // layer-2 buffers alias layer-1 buffers that are dead by then
  float*    h2     = h1;                 // N*64 <= N*256
  float*    asrc2  = asrc1;
  float*    adst2  = adst1;
  unsigned* mord2  = mord1;
  float*    denom2 = denom1;
  float*    outf   = (float*)d_out;

  const int T = 256;

  // ===== layer 1 =====
  {
    int mt64  = (Nn / 16 + 3) / 4;
    int tiles = mt64 * (C1 / 16);
    gemm_wmma_f32<<<cdiv(tiles, 8), 256, 0, stream>>>(x, W1, h1, Nn, C1, IN);
  }
  node_alpha<4, 64><<<cdiv((long long)Nn * 4, T), T, 0, stream>>>(
      h1, att_s1, att_d1, asrc1, adst1, Nn);
  fill_u32<<<cdiv((long long)Nn * 4, T), T, 0, stream>>>(mord1, 0u, (long long)Nn * 4);
  edge_max_k<4><<<cdiv((long long)Etot * 4, T), T, 0, stream>>>(
      ei, asrc1, adst1, mord1, Eraw, Etot);
  fill_f32<<<cdiv((long long)Nn * 4, T), T, 0, stream>>>(denom1, 0.f, (long long)Nn * 4);
  edge_expsum_k<4><<<cdiv((long long)Etot * 4, T), T, 0, stream>>>(
      ei, asrc1, adst1, mord1, denom1, Eraw, Etot);
  fill_f32<<<cdiv((long long)Nn * C1, T), T, 0, stream>>>(out1, 0.f, (long long)Nn * C1);
  edge_agg_k<4, 64><<<cdiv((long long)Etot * (C1 / 4), T), T, 0, stream>>>(
      ei, asrc1, adst1, mord1, denom1, h1, out1, Eraw, Etot);
  bias_elu<<<cdiv((long long)Nn * C1, T), T, 0, stream>>>(out1, b1, C1, (long long)Nn * C1);

  // ===== layer 2 =====
  {
    int mt64  = (Nn / 16 + 3) / 4;
    int tiles = mt64 * (F2 / 16);
    gemm_wmma_f32<<<cdiv(tiles, 8), 256, 0, stream>>>(out1, W2, h2, Nn, F2, C1);
  }
  node_alpha<1, 64><<<cdiv((long long)Nn, T), T, 0, stream>>>(
      h2, att_s2, att_d2, asrc2, adst2, Nn);
  fill_u32<<<cdiv((long long)Nn, T), T, 0, stream>>>(mord2, 0u, (long long)Nn);
  edge_max_k<1><<<cdiv((long long)Etot, T), T, 0, stream>>>(
      ei, asrc2, adst2, mord2, Eraw, Etot);
  fill_f32<<<cdiv((long long)Nn, T), T, 0, stream>>>(denom2, 0.f, (long long)Nn);
  edge_expsum_k<1><<<cdiv((long long)Etot, T), T, 0, stream>>>(
      ei, asrc2, adst2, mord2, denom2, Eraw, Etot);
  fill_f32<<<cdiv((long long)Nn * F2, T), T, 0, stream>>>(outf, 0.f, (long long)Nn * F2);
  edge_agg_k<1, 64><<<cdiv((long long)Etot * (F2 / 4), T), T, 0, stream>>>(
      ei, asrc2, adst2, mord2, denom2, h2, outf, Eraw, Etot);
  bias_add<<<cdiv((long long)Nn * F2, T), T, 0, stream>>>(outf, b2, F2, (long long)Nn * F2);
}